// ResGEChebNet_22995254903323
// MI455X (gfx1250) — compile-verified
//
#include <hip/hip_runtime.h>
#include <math.h>
#include <stdint.h>

#define NV    10000
#define NDEG  32
#define NB    16
#define NK    6
#define NE    (NV * NDEG)
#define BN_EPS 1e-5f

typedef float v2f __attribute__((ext_vector_type(2)));
typedef float v8f __attribute__((ext_vector_type(8)));
typedef unsigned int u32x4 __attribute__((ext_vector_type(4)));
typedef int i32x4 __attribute__((ext_vector_type(4)));
typedef int i32x8 __attribute__((ext_vector_type(8)));

// ---------------------------------------------------------------------------
// (B, C, V) -> (V, C, B)
// ---------------------------------------------------------------------------
__global__ void transpose_in_kernel(const float* __restrict__ x, float* __restrict__ z, int C) {
  size_t i = (size_t)blockIdx.x * blockDim.x + threadIdx.x;
  size_t N = (size_t)NV * C * NB;
  if (i >= N) return;
  int b = (int)(i % NB);
  size_t r = i / NB;
  int c = (int)(r % C);
  int v = (int)(r / C);
  z[i] = x[((size_t)b * C + c) * NV + v];
}

// ---------------------------------------------------------------------------
// Regular-degree SpMM: out[v,:] = alpha * sum_e val[v,e] * z[col[v,e],:]
//                                 (+ beta * y[v,:])
// Rows are (C*B) contiguous floats; edge list for node v is contiguous.
// The widest feature map (82 MB) fits in the 192 MB L2, so the 32-way gather
// is L2-resident; HBM sees ~read-once + write-once per SpMM.
// ---------------------------------------------------------------------------
__global__ void spmm_kernel(float* __restrict__ out, const float* __restrict__ z,
                            const float* __restrict__ y,
                            const int* __restrict__ col, const float* __restrict__ val,
                            int CB, float alpha, float beta) {
  __shared__ int   s_col[NDEG];
  __shared__ float s_val[NDEG];
  const int v = blockIdx.x;
  const int t = threadIdx.x;
  if (t < NDEG) {
    s_col[t] = col[v * NDEG + t];
    s_val[t] = val[v * NDEG + t];
  }
  __syncthreads();
  for (int j = t; j < CB; j += blockDim.x) {
    float acc = 0.0f;
#pragma unroll 8
    for (int e = 0; e < NDEG; ++e) {
      if ((e & 7) == 0 && (e + 8) < NDEG)
        __builtin_prefetch(&z[(size_t)s_col[e + 8] * CB + j], 0, 1);
      acc += s_val[e] * z[(size_t)s_col[e] * CB + j];
    }
    float r = alpha * acc;
    if (beta != 0.0f) r += beta * y[(size_t)v * CB + j];
    out[(size_t)v * CB + j] = r;
  }
}

// ---------------------------------------------------------------------------
// WMMA Chebyshev slice accumulate, weight tile staged via TDM.
// Block = 8 waves = 8 consecutive nodes, all sharing one 16-wide Cout tile.
// Wave 0 DMAs the (CIN x 16) weight tile into LDS with tensor_load_to_lds,
// waits on TENSORcnt, then the block barriers; every wave then runs
//   D(16b x 16o) (+)= A(16b x CIN) * W(CIN x 16o)
// fully unrolled over CIN in steps of 4 via v_wmma_f32_16x16x4_f32.
// Feature layout z: (V, CIN, B).  Output layout out: (V, Cout, B).
// flags: bit0 = init (zero C), bit1 = add bias[o], bit2 = relu at end.
// ---------------------------------------------------------------------------
template <int CIN>
__global__ void cheb_acc_kernel(float* __restrict__ out, const float* __restrict__ z,
                                const float* __restrict__ w, const float* __restrict__ bias,
                                int Cout, int flags) {
  __shared__ float s_w[CIN * 16];

  const int lane   = threadIdx.x & 31;
  const int waveId = threadIdx.x >> 5;
  const int tiles  = Cout >> 4;
  const int vblk   = blockIdx.x / tiles;       // uniform
  const int o0     = (blockIdx.x % tiles) << 4;
  const int v      = vblk * 8 + waveId;        // wave-uniform; NV % 8 == 0

  if (waveId == 0) {
    // ---- Tensor Data Mover: 2D tile (16 wide, CIN rows, row stride Cout) ----
    const uint64_t gaddr = (uint64_t)(uintptr_t)(w + (size_t)o0);
    const uint32_t laddr = (uint32_t)(uintptr_t)(&s_w[0]);
    const uint32_t td0 = (uint32_t)Cout;   // tensor_dim0
    const uint32_t td1 = (uint32_t)CIN;    // tensor_dim1
    const uint32_t tl0 = 16u;              // tile_dim0
    const uint32_t tl1 = (uint32_t)CIN;    // tile_dim1 (tile_dim2 = 0)
    const uint64_t st0 = (uint64_t)Cout;   // tensor_dim0_stride
    const uint64_t st1 = 0;                // tensor_dim1_stride (2D: unused)

    u32x4 g0;
    g0[0] = 1u;                                              // count=1, user mode
    g0[1] = laddr;                                           // lds_addr
    g0[2] = (uint32_t)(gaddr & 0xFFFFFFFFu);                 // global_addr lo
    g0[3] = (uint32_t)((gaddr >> 32) & 0x1FFFFFFu) | (2u << 30); // addr hi | type=2

    i32x8 g1;
    g1[0] = (int)(2u << 16);                                  // data_size=4B, mask=0
    g1[1] = (int)((td0 & 0xFFFFu) << 16);                     // tensor_dim0[15:0] @48
    g1[2] = (int)(((td0 >> 16) & 0xFFFFu) | ((td1 & 0xFFFFu) << 16));
    g1[3] = (int)(((td1 >> 16) & 0xFFFFu) | (tl0 << 16));     // tile_dim0 @112
    g1[4] = (int)(tl1);                                       // tile_dim1, tile_dim2=0
    g1[5] = (int)(uint32_t)(st0 & 0xFFFFFFFFu);               // stride0 lo
    g1[6] = (int)(((st0 >> 32) & 0xFFFFu) | ((st1 & 0xFFFFu) << 16));
    g1[7] = (int)(uint32_t)((st1 >> 16) & 0xFFFFFFFFu);       // stride1 hi

    i32x4 gz = {0, 0, 0, 0};
#if defined(__clang_major__) && (__clang_major__ >= 23)
    i32x8 gz8 = {0, 0, 0, 0, 0, 0, 0, 0};
    __builtin_amdgcn_tensor_load_to_lds(g0, g1, gz, gz, gz8, 0);
#else
    __builtin_amdgcn_tensor_load_to_lds(g0, g1, gz, gz, 0);
#endif
    __builtin_amdgcn_s_wait_tensorcnt(0);
  }
  __syncthreads();

  const int n  = lane & 15;
  const int hi = lane >> 4;

  const float* zrow = z + (size_t)v * CIN * NB;
  float* orow = out + (size_t)v * Cout * NB + (size_t)(o0 + n) * NB + hi * 8;

  v8f c = {};
  if (!(flags & 1)) {
#pragma unroll
    for (int j = 0; j < 8; ++j) c[j] = orow[j];
  }
  if (flags & 2) {
    float bv = bias[o0 + n];
#pragma unroll
    for (int j = 0; j < 8; ++j) c[j] += bv;
  }

#pragma unroll
  for (int kk = 0; kk < CIN; kk += 4) {
    const int k0 = kk + 2 * hi;      // lanes 0-15: K=kk,kk+1 ; lanes 16-31: K=kk+2,kk+3
    v2f a, bf;
    a.x  = zrow[(k0 + 0) * NB + n];  // A[M=n, K=k0]
    a.y  = zrow[(k0 + 1) * NB + n];  // A[M=n, K=k0+1]
    bf.x = s_w[(k0 + 0) * 16 + n];   // B[K=k0,   N=n]  (LDS, bank-conflict-free)
    bf.y = s_w[(k0 + 1) * 16 + n];   // B[K=k0+1, N=n]
    c = __builtin_amdgcn_wmma_f32_16x16x4_f32(false, a, false, bf,
                                              (short)0, c, false, false);
  }

  if (flags & 4) {
#pragma unroll
    for (int j = 0; j < 8; ++j) c[j] = fmaxf(c[j], 0.0f);
  }
#pragma unroll
  for (int j = 0; j < 8; ++j) orow[j] = c[j];
}

// ---------------------------------------------------------------------------
// BatchNorm stats: deterministic tree reduction, one block per channel.
// ---------------------------------------------------------------------------
__global__ void bn_stats_kernel(const float* __restrict__ z, float* __restrict__ sum,
                                float* __restrict__ sq, int C) {
  const int c = blockIdx.x;
  float s = 0.0f, q = 0.0f;
  for (int i = threadIdx.x; i < NV * NB; i += blockDim.x) {
    int v = i / NB, b = i % NB;
    float xv = z[((size_t)v * C + c) * NB + b];
    s += xv;
    q += xv * xv;
  }
  __shared__ float ss[256];
  __shared__ float qq[256];
  ss[threadIdx.x] = s;
  qq[threadIdx.x] = q;
  __syncthreads();
  for (int off = 128; off; off >>= 1) {
    if (threadIdx.x < off) {
      ss[threadIdx.x] += ss[threadIdx.x + off];
      qq[threadIdx.x] += qq[threadIdx.x + off];
    }
    __syncthreads();
  }
  if (threadIdx.x == 0) { sum[c] = ss[0]; sq[c] = qq[0]; }
}

__global__ void bn_finalize_kernel(const float* __restrict__ sum, const float* __restrict__ sq,
                                   const float* __restrict__ g, const float* __restrict__ b,
                                   float* __restrict__ scale, float* __restrict__ shift, int C) {
  int c = blockIdx.x * blockDim.x + threadIdx.x;
  if (c >= C) return;
  float n   = (float)NV * (float)NB;
  float m   = sum[c] / n;
  float var = sq[c] / n - m * m;
  float sc  = g[c] * rsqrtf(var + BN_EPS);
  scale[c] = sc;
  shift[c] = b[c] - m * sc;
}

__global__ void bn_apply_kernel(const float* __restrict__ zin, float* __restrict__ zout,
                                const float* __restrict__ scale, const float* __restrict__ shift,
                                int C) {
  size_t N = (size_t)NV * C * NB;
  for (size_t i = (size_t)blockIdx.x * blockDim.x + threadIdx.x; i < N;
       i += (size_t)gridDim.x * blockDim.x) {
    int c = (int)((i / NB) % C);
    zout[i] = zin[i] * scale[c] + shift[c];
  }
}

// ---------------------------------------------------------------------------
// Global max-pool over nodes: pooled[b, c] = max_v h[v, c, b]
// ---------------------------------------------------------------------------
__global__ void pool_max_kernel(const float* __restrict__ h, float* __restrict__ pooled, int C) {
  const int b = blockIdx.x / C;
  const int c = blockIdx.x % C;
  float m = -3.402823466e38f;
  for (int v = threadIdx.x; v < NV; v += blockDim.x)
    m = fmaxf(m, h[((size_t)v * C + c) * NB + b]);
  __shared__ float s[256];
  s[threadIdx.x] = m;
  __syncthreads();
  for (int off = 128; off; off >>= 1) {
    if (threadIdx.x < off) s[threadIdx.x] = fmaxf(s[threadIdx.x], s[threadIdx.x + off]);
    __syncthreads();
  }
  if (threadIdx.x == 0) pooled[b * C + c] = s[0];
}

// ---------------------------------------------------------------------------
// Head: BN over batch + linear(128->10) + relu + log_softmax.  One block.
// ---------------------------------------------------------------------------
__global__ void head_kernel(const float* __restrict__ pooled,
                            const float* __restrict__ g, const float* __restrict__ be,
                            const float* __restrict__ lw, const float* __restrict__ lb,
                            float* __restrict__ out) {
  const int C = 128, NO = 10;
  __shared__ float xn[NB * 128];
  __shared__ float logits[NB * 10];
  const int t = threadIdx.x;
  if (t < C) {
    float m = 0.0f, q = 0.0f;
    for (int b = 0; b < NB; ++b) {
      float x = pooled[b * C + t];
      m += x;
      q += x * x;
    }
    m /= (float)NB;
    float var = q / (float)NB - m * m;
    float sc  = g[t] * rsqrtf(var + BN_EPS);
    float sh  = be[t] - m * sc;
    for (int b = 0; b < NB; ++b) xn[b * C + t] = pooled[b * C + t] * sc + sh;
  }
  __syncthreads();
  if (t < NB * NO) {
    int b = t / NO, o = t % NO;
    float acc = lb[o];
    for (int c = 0; c < C; ++c) acc += xn[b * C + c] * lw[c * NO + o];
    logits[t] = fmaxf(acc, 0.0f);
  }
  __syncthreads();
  if (t < NB) {
    float mx = -3.402823466e38f;
    for (int o = 0; o < NO; ++o) mx = fmaxf(mx, logits[t * NO + o]);
    float s = 0.0f;
    for (int o = 0; o < NO; ++o) s += expf(logits[t * NO + o] - mx);
    float lse = mx + logf(s);
    for (int o = 0; o < NO; ++o) out[t * NO + o] = logits[t * NO + o] - lse;
  }
}

// ---------------------------------------------------------------------------
// Host orchestration
// ---------------------------------------------------------------------------
extern "C" void kernel_launch(void* const* d_in, const int* in_sizes, int n_in,
                              void* d_out, int out_size, void* d_ws, size_t ws_size,
                              hipStream_t stream) {
  (void)in_sizes; (void)n_in; (void)out_size; (void)ws_size;

  const float* x      = (const float*)d_in[0];
  const float* w_in   = (const float*)d_in[1];
  const float* b_in   = (const float*)d_in[2];
  const float* g1     = (const float*)d_in[3];
  const float* be1    = (const float*)d_in[4];
  const float* w1a    = (const float*)d_in[5];
  const float* b1a    = (const float*)d_in[6];
  const float* g1h    = (const float*)d_in[7];
  const float* be1h   = (const float*)d_in[8];
  const float* w1b    = (const float*)d_in[9];
  const float* b1b    = (const float*)d_in[10];
  const float* sc1    = (const float*)d_in[11];
  const float* g2     = (const float*)d_in[12];
  const float* be2    = (const float*)d_in[13];
  const float* w2a    = (const float*)d_in[14];
  const float* b2a    = (const float*)d_in[15];
  const float* g2h    = (const float*)d_in[16];
  const float* be2h   = (const float*)d_in[17];
  const float* w2b    = (const float*)d_in[18];
  const float* b2b    = (const float*)d_in[19];
  const float* sc2    = (const float*)d_in[20];
  const float* g_out  = (const float*)d_in[21];
  const float* be_out = (const float*)d_in[22];
  const float* lin_w  = (const float*)d_in[23];
  const float* lin_b  = (const float*)d_in[24];
  const float* lap_vals = (const float*)d_in[25];
  const int*   lap_idx  = (const int*)d_in[26];
  const int*   colp     = lap_idx + NE;   // lap_idx row 1 = cols

  float* ws = (float*)d_ws;
  const size_t SLOT = (size_t)NV * 128 * NB;
  float* S0 = ws;
  float* S1 = ws + 1 * SLOT;
  float* S2 = ws + 2 * SLOT;
  float* S3 = ws + 3 * SLOT;
  float* S4 = ws + 4 * SLOT;
  float* S5 = ws + 5 * SLOT;
  float* stat_sum   = ws + 6 * SLOT;
  float* stat_sq    = stat_sum + 128;
  float* stat_scale = stat_sq + 128;
  float* stat_shift = stat_scale + 128;
  float* pooled     = stat_shift + 128;     // NB*128

  auto launch_acc = [&](float* out, const float* zslice, const float* wk, const float* bias,
                        int Cin, int Cout, int flags) {
    dim3 g((NV / 8) * (Cout >> 4)), b(256);   // 8 wave32 = 8 nodes per block
    switch (Cin) {
      case 8:
        cheb_acc_kernel<8><<<g, b, 0, stream>>>(out, zslice, wk, bias, Cout, flags); break;
      case 32:
        cheb_acc_kernel<32><<<g, b, 0, stream>>>(out, zslice, wk, bias, Cout, flags); break;
      case 64:
        cheb_acc_kernel<64><<<g, b, 0, stream>>>(out, zslice, wk, bias, Cout, flags); break;
      default:
        cheb_acc_kernel<128><<<g, b, 0, stream>>>(out, zslice, wk, bias, Cout, flags); break;
    }
  };
  auto launch_spmm = [&](float* out, const float* z, const float* y, int CB, float a, float b) {
    spmm_kernel<<<dim3(NV), dim3(256), 0, stream>>>(out, z, y, colp, lap_vals, CB, a, b);
  };
  // Chebyshev layer: out (+)= sum_k T_k(L)x * w_k  (streams k, clobbers x0/t1/t2)
  auto run_cheb = [&](float* out, float* x0, const float* w, const float* bias,
                      int Cin, int Cout, bool relu, bool initOut, float* t1, float* t2) {
    launch_acc(out, x0, w, bias, Cin, Cout, (initOut ? 1 : 0) | (bias ? 2 : 0));
    launch_spmm(t1, x0, nullptr, Cin * NB, 1.0f, 0.0f);              // x1 = L x0
    launch_acc(out, t1, w + (size_t)1 * Cin * Cout, nullptr, Cin, Cout, 0);
    float* xm = x0; float* xc = t1; float* xn = t2;
    for (int k = 2; k < NK; ++k) {
      launch_spmm(xn, xc, xm, Cin * NB, 2.0f, -1.0f);                // x2 = 2 L x1 - x0
      launch_acc(out, xn, w + (size_t)k * Cin * Cout, nullptr, Cin, Cout,
                 (relu && k == NK - 1) ? 4 : 0);
      float* t = xm; xm = xc; xc = xn; xn = t;
    }
  };
  auto run_bn = [&](const float* zin, float* zout, const float* g, const float* b, int C) {
    bn_stats_kernel<<<dim3(C), dim3(256), 0, stream>>>(zin, stat_sum, stat_sq, C);
    bn_finalize_kernel<<<dim3(1), dim3(128), 0, stream>>>(stat_sum, stat_sq, g, b,
                                                          stat_scale, stat_shift, C);
    size_t N = (size_t)NV * C * NB;
    int blocks = (int)((N + 255) / 256);
    if (blocks > 2048) blocks = 2048;
    bn_apply_kernel<<<dim3(blocks), dim3(256), 0, stream>>>(zin, zout, stat_scale, stat_shift, C);
  };

  // ---- input layer: h0 = relu(cheb(x, w_in) + b_in), C 8 -> 32 ----
  {
    size_t N = (size_t)NV * 8 * NB;
    transpose_in_kernel<<<dim3((int)((N + 255) / 256)), dim3(256), 0, stream>>>(x, S1, 8);
  }
  run_cheb(S0, S1, w_in, b_in, 8, 32, true, true, S3, S4);

  // ---- block 1: 32 -> 64 ----
  run_bn(S0, S1, g1, be1, 32);                       // XB = bn(h)      (S1)
  launch_acc(S5, S1, sc1, nullptr, 32, 64, 1);       // OUT = shortcut  (init)
  run_cheb(S2, S1, w1a, b1a, 32, 64, true, true, S3, S4);  // O1 = relu(chebA)
  run_bn(S2, S2, g1h, be1h, 64);                     // O1 = bn(O1) in place
  run_cheb(S5, S2, w1b, b1b, 64, 64, true, false, S3, S4); // OUT = relu(OUT + bias + chebB)

  // ---- block 2: 64 -> 128 ----
  run_bn(S5, S1, g2, be2, 64);
  launch_acc(S0, S1, sc2, nullptr, 64, 128, 1);
  run_cheb(S2, S1, w2a, b2a, 64, 128, true, true, S3, S4);
  run_bn(S2, S2, g2h, be2h, 128);
  run_cheb(S0, S2, w2b, b2b, 128, 128, true, false, S3, S4);

  // ---- head ----
  pool_max_kernel<<<dim3(NB * 128), dim3(256), 0, stream>>>(S0, pooled, 128);
  head_kernel<<<dim3(1), dim3(256), 0, stream>>>(pooled, g_out, be_out, lin_w, lin_b,
                                                 (float*)d_out);
}